// TilesDoyPositionEncoder_35502199668788
// MI455X (gfx1250) — compile-verified
//
#include <hip/hip_runtime.h>
#include <hip/hip_bf16.h>
#include <stdint.h>

// out[row, :] = W[days[row], :] + bias[:]   (D = 1024, 365-row table)
//
// Roofline: ~0 FLOPs, 512MB fp32 output stream -> HBM store-bound (~22us at
// 23.3 TB/s). W (1.5MB) is L2-resident. CDNA5 path: TDM gather mode
// (tensor_load_to_lds, TENSORcnt) performs the indexed row-gather in the DMA
// engine; waves only do LDS -> +bias -> non-temporal b128 stores.

typedef float    v4f   __attribute__((ext_vector_type(4)));
typedef uint32_t u32;
typedef u32      su32x4 __attribute__((ext_vector_type(4)));
typedef u32      su32x8 __attribute__((ext_vector_type(8)));

#define TILE_ROWS 8            // rows per TDM gather (16-bit index mode)
#define ROWS_PER_BLOCK 32      // 4 tiles per block

// Issue one TDM gather of `valid` rows (indices packed 16-bit, 2/dword) of a
// [n_days x D] fp32 tensor at wbase into LDS at lds_byte_addr.
__device__ __forceinline__ void tdm_gather_rows(u32 lds_byte_addr, uint64_t wbase,
                                                u32 D, u32 n_days, u32 valid,
                                                u32 i01, u32 i23, u32 i45, u32 i67) {
    su32x4 g0;
    g0.x = 0x80000001u;                                   // gather_mode=1, 16b idx, count=1
    g0.y = lds_byte_addr;                                 // D#.lds_addr
    g0.z = (u32)wbase;                                    // global_addr[31:0]
    g0.w = ((u32)(wbase >> 32) & 0x01FFFFFFu) | 0x80000000u; // addr[56:32] | type=2

    su32x8 g1;
    g1.s0 = 0x00020000u;                                  // data_size=2 (4B), mask=0, no pad
    g1.s1 = (D & 0xFFFFu) << 16;                          // atomic_bar=0 | tensor_dim0 lo
    g1.s2 = ((D >> 16) & 0xFFFFu) | ((n_days & 0xFFFFu) << 16); // dim0 hi | tensor_dim1 lo
    g1.s3 = ((n_days >> 16) & 0xFFFFu) | ((D & 0xFFFFu) << 16); // dim1 hi | tile_dim0 = D
    g1.s4 = valid & 0xFFFFu;                              // tile_dim1 = #indices, tile_dim2=0
    g1.s5 = D;                                            // tensor_dim0_stride lo
    g1.s6 = 0u;                                           // stride hi | dim1_stride (ignored)
    g1.s7 = 0u;

    // Index dwords come from global loads (divergent-typed to the compiler
    // even though block-uniform); readfirstlane makes them SGPR-legal for the
    // "s" asm constraint (clang-22 won't insert the copy itself).
    su32x4 g2;                                            // row_index_0..7 (16-bit packed)
    g2.x = (u32)__builtin_amdgcn_readfirstlane((int)i01);
    g2.y = (u32)__builtin_amdgcn_readfirstlane((int)i23);
    g2.z = (u32)__builtin_amdgcn_readfirstlane((int)i45);
    g2.w = (u32)__builtin_amdgcn_readfirstlane((int)i67);
    su32x4 g3;                                            // row_index_8..15: unused
    g3.x = 0u; g3.y = 0u; g3.z = 0u; g3.w = 0u;

    asm volatile("tensor_load_to_lds %0, %1, %2, %3"
                 :
                 : "s"(g0), "s"(g1), "s"(g2), "s"(g3)
                 : "memory");
}

__global__ void __launch_bounds__(256)
doy_gather_bias_kernel(const int* __restrict__ days,
                       const float* __restrict__ W,
                       const float* __restrict__ bias,
                       float* __restrict__ out,
                       int n_rows, int D, int n_days) {
    const int tid  = threadIdx.x;                         // 0 .. D/4-1 (256)
    const int row0 = blockIdx.x * ROWS_PER_BLOCK;
    const int nr   = min(ROWS_PER_BLOCK, n_rows - row0);
    if (nr <= 0) return;
    const int ntiles = (nr + TILE_ROWS - 1) / TILE_ROWS;

    // Bias slice lives in registers for the whole block (b128 load).
    const v4f b4 = reinterpret_cast<const v4f*>(bias)[tid];

    // Double-buffered gather target: 2 x (8 rows x 4KB) = 64KB LDS.
    __shared__ __align__(16) float smem[2 * TILE_ROWS * 1024];
    const u32 lds_base = (u32)(uintptr_t)(&smem[0]);      // low 32 bits = LDS offset
    const u32 buf_b    = (u32)(TILE_ROWS * D * 4);        // bytes per buffer

    // TDM ignores EXEC, so the issuer guard must be a *scalar* branch:
    // readfirstlane(tid) is wave-uniform -> s_cmp/s_cbranch, waves 1..7 jump over.
    const bool issuer = (__builtin_amdgcn_readfirstlane((int)threadIdx.x) >> 5) == 0;
    const uint64_t wbase = (uint64_t)(uintptr_t)W;

    if (issuer) {
        // Prologue: gather tile 0 into buffer 0.
        const int* ip = days + row0;
        u32 d0=(u32)ip[0], d1=(u32)ip[1], d2=(u32)ip[2], d3=(u32)ip[3];
        u32 d4=(u32)ip[4], d5=(u32)ip[5], d6=(u32)ip[6], d7=(u32)ip[7];
        tdm_gather_rows(lds_base, wbase, (u32)D, (u32)n_days,
                        (u32)min(TILE_ROWS, nr),
                        (d0 & 0xFFFFu) | (d1 << 16), (d2 & 0xFFFFu) | (d3 << 16),
                        (d4 & 0xFFFFu) | (d5 << 16), (d6 & 0xFFFFu) | (d7 << 16));
    }

    for (int t = 0; t < ntiles; ++t) {
        const int cur = t & 1;
        if (issuer) {
            if (t + 1 < ntiles) {
                // Gather tile t+1 into the other buffer, then wait until only
                // it is outstanding -> tile t complete (TDM in-order per wave).
                const int* ip = days + row0 + (t + 1) * TILE_ROWS;
                u32 d0=(u32)ip[0], d1=(u32)ip[1], d2=(u32)ip[2], d3=(u32)ip[3];
                u32 d4=(u32)ip[4], d5=(u32)ip[5], d6=(u32)ip[6], d7=(u32)ip[7];
                tdm_gather_rows(lds_base + (u32)(cur ^ 1) * buf_b, wbase,
                                (u32)D, (u32)n_days,
                                (u32)min(TILE_ROWS, nr - (t + 1) * TILE_ROWS),
                                (d0 & 0xFFFFu) | (d1 << 16), (d2 & 0xFFFFu) | (d3 << 16),
                                (d4 & 0xFFFFu) | (d5 << 16), (d6 & 0xFFFFu) | (d7 << 16));
                __builtin_amdgcn_s_wait_tensorcnt(1);
            } else {
                __builtin_amdgcn_s_wait_tensorcnt(0);
            }
        }
        __syncthreads();   // publish buffer `cur` to all 8 waves

        // Stream 8 rows: LDS b128 -> +bias -> non-temporal b128 store.
        const float* src = &smem[cur * TILE_ROWS * D];
#pragma unroll
        for (int j = 0; j < TILE_ROWS; ++j) {
            const int r = t * TILE_ROWS + j;
            if (r < nr) {
                const v4f v = *reinterpret_cast<const v4f*>(src + j * D + tid * 4);
                const v4f o = v + b4;
                v4f* dst = reinterpret_cast<v4f*>(out + (size_t)(row0 + r) * (size_t)D) + tid;
                __builtin_nontemporal_store(o, dst);
            }
        }
        __syncthreads();   // retire buffer `cur` before TDM overwrites it
    }
}

extern "C" void kernel_launch(void* const* d_in, const int* in_sizes, int n_in,
                              void* d_out, int out_size, void* d_ws, size_t ws_size,
                              hipStream_t stream) {
    const int*   days = (const int*)d_in[0];    // [B*T] int32
    const float* W    = (const float*)d_in[1];  // [n_days, D] fp32
    const float* bias = (const float*)d_in[2];  // [D] fp32
    float*       out  = (float*)d_out;          // [B*T, D] fp32

    const int n_rows = in_sizes[0];             // B*T = 131072
    const int D      = in_sizes[2];             // 1024
    const int n_days = in_sizes[1] / D;         // 365
    const int blocks = (n_rows + ROWS_PER_BLOCK - 1) / ROWS_PER_BLOCK; // 4096

    dim3 grid(blocks), block(D / 4);            // 256 threads = 8 wave32
    hipLaunchKernelGGL(doy_gather_bias_kernel, grid, block, 0, stream,
                       days, W, bias, out, n_rows, D, n_days);
}